// HeteroForecastSageConv_85822036509291
// MI455X (gfx1250) — compile-verified
//
#include <hip/hip_runtime.h>
#include <math.h>

#define Hdim 64
#define T_STEPS 24
#define NSTORE 20000
#define NREGION 2000
#define E_SS 640000
#define E_RS 160000
#define ALPHA_C 0.5f
#define WAVES_PB 4

typedef float v2f __attribute__((ext_vector_type(2)));
typedef float v8f __attribute__((ext_vector_type(8)));
typedef unsigned int u32x4 __attribute__((ext_vector_type(4)));
typedef int i32x4 __attribute__((ext_vector_type(4)));
typedef int i32x8 __attribute__((ext_vector_type(8)));

__device__ __forceinline__ v8f wmma_f32(v2f a, v2f b, v8f c) {
    // D = A(16x4,f32) * B(4x16,f32) + C(16x16,f32)
    return __builtin_amdgcn_wmma_f32_16x16x4_f32(false, a, false, b, (short)0, c, false, false);
}

__device__ __forceinline__ float fast_rcp(float x) {
#if defined(__has_builtin)
#if __has_builtin(__builtin_amdgcn_rcpf)
    return __builtin_amdgcn_rcpf(x);
#else
    return 1.0f / x;
#endif
#else
    return 1.0f / x;
#endif
}
__device__ __forceinline__ float sigf(float x) { return fast_rcp(1.0f + __expf(-x)); }
__device__ __forceinline__ float tanh_fast(float x) {
    // 1 - 2/(e^{2x}+1); exp->inf gives rcp->0 => +1, exp->0 gives -1 (correct saturation)
    return 1.0f - 2.0f * fast_rcp(__expf(2.0f * x) + 1.0f);
}

// ---------------------------------------------------------------------------
// LSTM over T steps; each wave owns 16 nodes; gates = x_t*w_in + h@Whh^T + b.
// Dynamic LDS layout (floats):
//   [0,16384)          Whh (256x64 row-major, = B^T fragments, k contiguous)
//   [16384,16640)      w_in (256)
//   [16640,16896)      bias (256) = bih+bhh
//   [16896, +W*2048)   per-wave h double buffer (2 x 16 x 64)
//   then per-wave x stage (24 x 16)
// ---------------------------------------------------------------------------
__global__ __launch_bounds__(WAVES_PB * 32)
void lstm_kernel(const float* __restrict__ x,    // (N, T)
                 const float* __restrict__ Wih,  // (256,1)
                 const float* __restrict__ Whh,  // (256,64)
                 const float* __restrict__ bih,  // (256)
                 const float* __restrict__ bhh,  // (256)
                 float* __restrict__ hout,       // (N, 64)
                 int N)
{
    extern __shared__ float lds[];
    float* whh  = lds;
    float* win  = lds + 16384;
    float* bb   = lds + 16640;
    float* hbuf = lds + 16896;
    float* xbuf = lds + 16896 + WAVES_PB * 2048;

    const int tid  = threadIdx.x;
    const int wave = tid >> 5;
    const int lane = tid & 31;

    // --- Stage Whh (64 KB contiguous) into LDS ---------------------------
#if defined(__has_builtin) && __has_builtin(__builtin_amdgcn_tensor_load_to_lds)
    if (wave == 0) {
        // Tensor DMA descriptor (D#): 1D copy of 16384 f32 elements.
        unsigned long long ga = (unsigned long long)(uintptr_t)Whh;
        unsigned int ldsaddr  = (unsigned int)(uintptr_t)whh;  // low 32 bits = LDS byte offset
        u32x4 g0;
        g0.x = 0x1u;                                   // count=1, user mode
        g0.y = ldsaddr;                                // lds_addr
        g0.z = (unsigned int)(ga & 0xFFFFFFFFu);       // global_addr[31:0]
        g0.w = (unsigned int)((ga >> 32) & 0x01FFFFFFu) | (2u << 30); // addr[56:32], type=2
        i32x8 g1 = (i32x8){0,0,0,0,0,0,0,0};
        g1[0] = (int)(2u << 16);                       // data_size=4B, wg_mask=0
        g1[1] = (int)(16384u << 16);                   // tensor_dim0[15:0] -> bits[63:48]
        g1[2] = (int)((16384u >> 16) | (1u << 16));    // tensor_dim0 hi | tensor_dim1=1 lo
        g1[3] = (int)(16384u << 16);                   // tensor_dim1 hi=0 | tile_dim0=16384
        g1[4] = 1;                                     // tile_dim1=1, tile_dim2=0
        g1[5] = 16384;                                 // tensor_dim0_stride lo32
        g1[6] = 0;                                     // stride0 hi | stride1 lo
        g1[7] = 0;                                     // stride1 hi
        i32x4 z4 = (i32x4){0,0,0,0};
#if __has_include(<hip/amd_detail/amd_gfx1250_TDM.h>)
        i32x8 z8 = (i32x8){0,0,0,0,0,0,0,0};
        __builtin_amdgcn_tensor_load_to_lds(g0, g1, z4, z4, z8, 0);
#else
        __builtin_amdgcn_tensor_load_to_lds(g0, g1, z4, z4, 0);
#endif
        __builtin_amdgcn_s_wait_tensorcnt(0);
    }
#else
    for (int i = tid; i < 256 * 64; i += blockDim.x) whh[i] = Whh[i];
#endif
    for (int i = tid; i < 256; i += blockDim.x) {
        win[i] = Wih[i];
        bb[i]  = bih[i] + bhh[i];
    }

    const int hi   = lane >> 4;   // 0/1 : selects K sub-pair / M+8 rows
    const int lo   = lane & 15;
    const int gbase = (blockIdx.x * WAVES_PB + wave) * 16;
    float* hw = hbuf + wave * 2048;
    float* xw = xbuf + wave * (T_STEPS * 16);

    const bool active = (gbase < N);
    if (active) {
        for (int i = lane; i < T_STEPS * 16; i += 32) {
            int t = i >> 4, m = i & 15;
            xw[t * 16 + m] = x[(gbase + m) * T_STEPS + t];
        }
        for (int i = lane; i < 1024; i += 32) hw[i] = 0.0f;  // h buffer 0
    }
    __syncthreads();
    if (!active) return;

    v8f cmat[4];
    #pragma unroll
    for (int j = 0; j < 4; ++j) cmat[j] = (v8f){0,0,0,0,0,0,0,0};

    int cur = 0;
    #pragma unroll 1
    for (int t = 0; t < T_STEPS; ++t) {
        float* hcur = hw + cur * 1024;
        float* hnxt = hw + (cur ^ 1) * 1024;

        // A fragments of h (shared by all 16 output tiles this step)
        v2f afrag[16];
        #pragma unroll
        for (int kt = 0; kt < 16; ++kt)
            afrag[kt] = *(const v2f*)(hcur + lo * 64 + kt * 4 + 2 * hi);

        // x values for this lane's 8 rows (m = 8*hi + v)
        float4 xa = *(const float4*)(xw + t * 16 + 8 * hi);
        float4 xb = *(const float4*)(xw + t * 16 + 8 * hi + 4);
        float xv[8] = {xa.x, xa.y, xa.z, xa.w, xb.x, xb.y, xb.z, xb.w};

        #pragma unroll
        for (int jt = 0; jt < 4; ++jt) {          // feature column tile of h
            v8f g[4];
            #pragma unroll
            for (int q = 0; q < 4; ++q) {         // i, f, g, o
                v8f acc = (v8f){0,0,0,0,0,0,0,0};
                const int n = q * 64 + jt * 16 + lo;   // gate column
                #pragma unroll
                for (int kt = 0; kt < 16; ++kt) {
                    v2f bfrag = *(const v2f*)(whh + n * 64 + kt * 4 + 2 * hi);
                    acc = wmma_f32(afrag[kt], bfrag, acc);
                }
                const float wn = win[n], bn = bb[n];
                #pragma unroll
                for (int v = 0; v < 8; ++v) acc[v] += xv[v] * wn + bn;
                g[q] = acc;
            }
            #pragma unroll
            for (int v = 0; v < 8; ++v) {
                float ig = sigf(g[0][v]);
                float fg = sigf(g[1][v]);
                float gg = tanh_fast(g[2][v]);
                float og = sigf(g[3][v]);
                float cn = fg * cmat[jt][v] + ig * gg;
                cmat[jt][v] = cn;
                int m = v + 8 * hi;
                hnxt[m * 64 + jt * 16 + lo] = og * tanh_fast(cn);
            }
        }
        cur ^= 1;
    }

    float* hfin = hw + cur * 1024;
    for (int i = lane; i < 1024; i += 32)
        hout[gbase * 64 + i] = hfin[i];
}

// ---------------------------------------------------------------------------
__global__ void zero_f32(float* __restrict__ p, int n) {
    int t = blockIdx.x * blockDim.x + threadIdx.x;
    if (t < n) p[t] = 0.0f;
}

// 16 threads per edge; each handles a float4 chunk; both directions of ss graph
__global__ void scatter_ss(const int* __restrict__ edge_ss, const float* __restrict__ hs,
                           float* __restrict__ accF, float* __restrict__ accR,
                           float* __restrict__ cntF, float* __restrict__ cntR, int E) {
    int t = blockIdx.x * blockDim.x + threadIdx.x;
    int e = t >> 4, c = (t & 15) * 4;
    if (e >= E) return;
    int s = edge_ss[e];
    int d = edge_ss[E + e];
    float4 vs = *(const float4*)(hs + s * 64 + c);
    float4 vd = *(const float4*)(hs + d * 64 + c);
    float* pF = accF + d * 64 + c;
    float* pR = accR + s * 64 + c;
    atomicAdd(pF + 0, vs.x); atomicAdd(pF + 1, vs.y);
    atomicAdd(pF + 2, vs.z); atomicAdd(pF + 3, vs.w);
    atomicAdd(pR + 0, vd.x); atomicAdd(pR + 1, vd.y);
    atomicAdd(pR + 2, vd.z); atomicAdd(pR + 3, vd.w);
    if (c == 0) {
        atomicAdd(&cntF[d], 1.0f);
        atomicAdd(&cntR[s], 1.0f);
    }
}

__global__ void scatter_rs(const int* __restrict__ src, const int* __restrict__ dst,
                           const float* __restrict__ hr,
                           float* __restrict__ accRS, float* __restrict__ cntRS, int E) {
    int t = blockIdx.x * blockDim.x + threadIdx.x;
    int e = t >> 4, c = (t & 15) * 4;
    if (e >= E) return;
    int s = src[e], d = dst[e];
    float4 v = *(const float4*)(hr + s * 64 + c);
    float* p = accRS + d * 64 + c;
    atomicAdd(p + 0, v.x); atomicAdd(p + 1, v.y);
    atomicAdd(p + 2, v.z); atomicAdd(p + 3, v.w);
    if (c == 0) atomicAdd(&cntRS[d], 1.0f);
}

__global__ void finalize_means(float* __restrict__ accF, float* __restrict__ accR,
                               float* __restrict__ accRS,
                               const float* __restrict__ cF, const float* __restrict__ cR,
                               const float* __restrict__ cRS, int N) {
    int t = blockIdx.x * blockDim.x + threadIdx.x;
    if (t >= N * 64) return;
    int i = t >> 6;
    accF[t]  /= fmaxf(cF[i], 1.0f);
    accR[t]  /= fmaxf(cR[i], 1.0f);
    accRS[t] /= fmaxf(cRS[i], 1.0f);
}

// ---------------------------------------------------------------------------
// Fused per-store epilogue:
// u = relu(0.5*(hs@(Wself)^T + hs@(Wr_rs)^T + (1-a)mF@Ws2d^T + a*mR@Wd2s^T
//              + mRS@Wl_rs^T + bcomb) + hs);  out = u@Wf^T + bf
// ---------------------------------------------------------------------------
__global__ __launch_bounds__(WAVES_PB * 32)
void fused_store_out(const float* __restrict__ hs, const float* __restrict__ mF,
                     const float* __restrict__ mR, const float* __restrict__ mRS,
                     const float* __restrict__ W_s2d, const float* __restrict__ b_s2d,
                     const float* __restrict__ W_d2s, const float* __restrict__ b_d2s,
                     const float* __restrict__ W_self, const float* __restrict__ b_self,
                     const float* __restrict__ Wl_rs, const float* __restrict__ bl_rs,
                     const float* __restrict__ Wr_rs, const float* __restrict__ Wf,
                     const float* __restrict__ bf, float* __restrict__ out, int N)
{
    extern __shared__ float lds[];
    float* wSelf = lds;
    float* wS2d  = lds + 4096;
    float* wD2s  = lds + 8192;
    float* wLrs  = lds + 12288;
    float* wRrs  = lds + 16384;
    float* wF    = lds + 20480;
    float* bcomb = lds + 24576;
    float* bfl   = lds + 24640;
    float* ubuf0 = lds + 24704;

    const int tid = threadIdx.x;
    for (int i = tid; i < 4096; i += blockDim.x) {
        wSelf[i] = W_self[i]; wS2d[i] = W_s2d[i]; wD2s[i] = W_d2s[i];
        wLrs[i]  = Wl_rs[i];  wRrs[i] = Wr_rs[i]; wF[i]   = Wf[i];
    }
    if (tid < 64) {
        bcomb[tid] = b_self[tid] + (1.0f - ALPHA_C) * b_s2d[tid]
                   + ALPHA_C * b_d2s[tid] + bl_rs[tid];
        bfl[tid] = bf[tid];
    }
    __syncthreads();

    const int wave = tid >> 5;
    const int lane = tid & 31;
    const int hi = lane >> 4, lo = lane & 15;
    const int gbase = (blockIdx.x * WAVES_PB + wave) * 16;
    if (gbase >= N) return;
    float* ubuf = ubuf0 + wave * 1024;

    const float* rowh  = hs  + (long)(gbase + lo) * 64;
    const float* rowf  = mF  + (long)(gbase + lo) * 64;
    const float* rowr  = mR  + (long)(gbase + lo) * 64;
    const float* rowrs = mRS + (long)(gbase + lo) * 64;

    #pragma unroll
    for (int nt = 0; nt < 4; ++nt) {
        v8f acc = (v8f){0,0,0,0,0,0,0,0};
        const int n = nt * 16 + lo;
        #pragma unroll
        for (int kt = 0; kt < 16; ++kt) {
            const int k = kt * 4 + 2 * hi;
            v2f a_h  = *(const v2f*)(rowh + k);
            v2f a_f  = *(const v2f*)(rowf + k)  * (1.0f - ALPHA_C);
            v2f a_r  = *(const v2f*)(rowr + k)  * ALPHA_C;
            v2f a_rs = *(const v2f*)(rowrs + k);
            acc = wmma_f32(a_h,  *(const v2f*)(wSelf + n * 64 + k), acc);
            acc = wmma_f32(a_h,  *(const v2f*)(wRrs  + n * 64 + k), acc);
            acc = wmma_f32(a_f,  *(const v2f*)(wS2d  + n * 64 + k), acc);
            acc = wmma_f32(a_r,  *(const v2f*)(wD2s  + n * 64 + k), acc);
            acc = wmma_f32(a_rs, *(const v2f*)(wLrs  + n * 64 + k), acc);
        }
        const float bc = bcomb[n];
        #pragma unroll
        for (int v = 0; v < 8; ++v) {
            int m = v + 8 * hi;
            float hv = hs[(long)(gbase + m) * 64 + n];
            float val = 0.5f * (acc[v] + bc) + hv;
            ubuf[m * 64 + n] = fmaxf(val, 0.0f);
        }
    }

    // second GEMM: out = u @ Wf^T + bf  (u re-read in A layout from LDS)
    #pragma unroll
    for (int nt = 0; nt < 4; ++nt) {
        v8f acc = (v8f){0,0,0,0,0,0,0,0};
        const int n = nt * 16 + lo;
        #pragma unroll
        for (int kt = 0; kt < 16; ++kt) {
            const int k = kt * 4 + 2 * hi;
            v2f a_u = *(const v2f*)(ubuf + lo * 64 + k);
            acc = wmma_f32(a_u, *(const v2f*)(wF + n * 64 + k), acc);
        }
        const float bn = bfl[n];
        #pragma unroll
        for (int v = 0; v < 8; ++v) {
            int m = v + 8 * hi;
            out[(long)(gbase + m) * 64 + n] = acc[v] + bn;
        }
    }
}

// ---------------------------------------------------------------------------
extern "C" void kernel_launch(void* const* d_in, const int* in_sizes, int n_in,
                              void* d_out, int out_size, void* d_ws, size_t ws_size,
                              hipStream_t stream) {
    const float* x_store  = (const float*)d_in[0];
    const float* x_region = (const float*)d_in[1];
    const int*   edge_ss  = (const int*)d_in[2];
    const int*   e_rs_src = (const int*)d_in[3];
    const int*   e_rs_dst = (const int*)d_in[4];
    // d_in[5], d_in[6] (edge_sr_*) unused: region_h never reaches the output
    const float* Wih_s = (const float*)d_in[7];
    const float* Whh_s = (const float*)d_in[8];
    const float* bih_s = (const float*)d_in[9];
    const float* bhh_s = (const float*)d_in[10];
    const float* Wih_r = (const float*)d_in[11];
    const float* Whh_r = (const float*)d_in[12];
    const float* bih_r = (const float*)d_in[13];
    const float* bhh_r = (const float*)d_in[14];
    const float* W_s2d = (const float*)d_in[15];
    const float* b_s2d = (const float*)d_in[16];
    const float* W_d2s = (const float*)d_in[17];
    const float* b_d2s = (const float*)d_in[18];
    const float* W_self = (const float*)d_in[19];
    const float* b_self = (const float*)d_in[20];
    const float* Wl_rs  = (const float*)d_in[21];
    const float* bl_rs  = (const float*)d_in[22];
    const float* Wr_rs  = (const float*)d_in[23];
    // d_in[24..26] (Wl_sr, bl_sr, Wr_sr) unused
    const float* Wf = (const float*)d_in[27];
    const float* bf = (const float*)d_in[28];
    float* outp = (float*)d_out;

    float* ws    = (float*)d_ws;
    float* hs    = ws;                      // 20000*64
    float* hr    = hs    + NSTORE * 64;     // 2000*64
    float* accF  = hr    + NREGION * 64;    // 20000*64
    float* accR  = accF  + NSTORE * 64;
    float* accRS = accR  + NSTORE * 64;
    float* cntF  = accRS + NSTORE * 64;     // 20000
    float* cntR  = cntF  + NSTORE;
    float* cntRS = cntR  + NSTORE;

    const int nz = 3 * NSTORE * 64 + 3 * NSTORE;  // acc* + cnt* contiguous
    zero_f32<<<(nz + 255) / 256, 256, 0, stream>>>(accF, nz);

    const size_t lstm_lds = (size_t)(16896 + WAVES_PB * 2048 + WAVES_PB * T_STEPS * 16) * sizeof(float);
    lstm_kernel<<<(NSTORE / 16 + WAVES_PB - 1) / WAVES_PB, WAVES_PB * 32, lstm_lds, stream>>>(
        x_store, Wih_s, Whh_s, bih_s, bhh_s, hs, NSTORE);
    lstm_kernel<<<(NREGION / 16 + WAVES_PB - 1) / WAVES_PB, WAVES_PB * 32, lstm_lds, stream>>>(
        x_region, Wih_r, Whh_r, bih_r, bhh_r, hr, NREGION);

    scatter_ss<<<(E_SS * 16) / 256, 256, 0, stream>>>(edge_ss, hs, accF, accR, cntF, cntR, E_SS);
    scatter_rs<<<(E_RS * 16) / 256, 256, 0, stream>>>(e_rs_src, e_rs_dst, hr, accRS, cntRS, E_RS);
    finalize_means<<<(NSTORE * 64 + 255) / 256, 256, 0, stream>>>(accF, accR, accRS,
                                                                  cntF, cntR, cntRS, NSTORE);

    const size_t fused_lds = (size_t)(24704 + WAVES_PB * 1024) * sizeof(float);
    fused_store_out<<<(NSTORE / 16 + WAVES_PB - 1) / WAVES_PB, WAVES_PB * 32, fused_lds, stream>>>(
        hs, accF, accR, accRS, W_s2d, b_s2d, W_d2s, b_d2s, W_self, b_self,
        Wl_rs, bl_rs, Wr_rs, Wf, bf, outp, NSTORE);
}